// PitchLoss_27900107555524
// MI455X (gfx1250) — compile-verified
//
#include <hip/hip_runtime.h>

// ---------------------------------------------------------------------------
// PitchLoss on MI455X (gfx1250).
//   numerator  = sum_{non-pad elems} [max(p,0) + log1p(exp(-|p|))]
//              - sum_{non-pad rows}  sum_i k_i * p[reflected tap i of q]
//   denom      = 50 * (#non-pad rows)
// 105 MB streamed once via global_load_async_to_lds_b128 (ASYNCcnt DMA),
// tile reused twice out of LDS. Deterministic 2-kernel reduction; the
// 32->16 cross-lane step uses V_WMMA_F32_16X16X4_F32 (f32, lossless).
// ---------------------------------------------------------------------------

typedef float v2f __attribute__((ext_vector_type(2)));
typedef float v8f __attribute__((ext_vector_type(8)));

#define NBINS   50
#define TROWS   64
#define TELEMS  (TROWS * NBINS)   // 3200 floats / tile (12.8 KB)
#define BLOCK   256
#define MAXBLK  4096

// Gaussian taps: kernel_size=5, sigma=0.5 -> pdf = [e^-8, e^-2, 1, e^-2, e^-8]/S
#define K0 2.6386462e-04f
#define K1 1.0645079e-01f
#define K2 7.8657076e-01f

// Deterministic block reduction of (num, cnt). The 32->16 stage is done with
// V_WMMA_F32_16X16X4_F32: A(16x4) is a K-selector of ones, B(4x16) holds the
// 32 surviving lane values (vgpr0 = rows K=0,2 -> num; vgpr1 = rows K=1,3 ->
// cnt). D[m][n] = sum_k A[m][k]*B[k][n] gives colsum(n) = v[n] + v[n+16] in
// every lane's d[0]. Executed by ALL threads (EXEC must be all ones for WMMA).
__device__ __forceinline__ float2 block_reduce_pair(float accN, float accC,
                                                    float* sNum, float* sCnt) {
  const int tid = threadIdx.x;
  sNum[tid] = accN;
  sCnt[tid] = accC;
  __syncthreads();
#pragma unroll
  for (int s = BLOCK / 2; s >= 32; s >>= 1) {
    if (tid < s) { sNum[tid] += sNum[tid + s]; sCnt[tid] += sCnt[tid + s]; }
    __syncthreads();
  }
  float bn = sNum[tid & 31];
  float bc = sCnt[tid & 31];
  v2f b;  b.x  = bn;   b.y  = bc;
  v2f aN; aN.x = 1.0f; aN.y = 0.0f;   // select K rows {0,2} -> numerator
  v2f aC; aC.x = 0.0f; aC.y = 1.0f;   // select K rows {1,3} -> count
  v8f z = {};
  v8f dn = __builtin_amdgcn_wmma_f32_16x16x4_f32(false, aN, false, b,
                                                 (short)0, z, false, false);
  v8f dc = __builtin_amdgcn_wmma_f32_16x16x4_f32(false, aC, false, b,
                                                 (short)0, z, false, false);
  __syncthreads();
  if (tid < 16) { sNum[tid] = dn[0]; sCnt[tid] = dc[0]; }
  __syncthreads();
#pragma unroll
  for (int s = 8; s >= 1; s >>= 1) {
    if (tid < s) { sNum[tid] += sNum[tid + s]; sCnt[tid] += sCnt[tid + s]; }
    __syncthreads();
  }
  float2 r; r.x = sNum[0]; r.y = sCnt[0];
  return r;
}

__global__ __launch_bounds__(BLOCK)
void pitch_partial_kernel(const float* __restrict__ preds,
                          const int* __restrict__ gt,
                          float* __restrict__ partials,
                          int rows) {
  __shared__ __align__(16) float tile[TELEMS];
  __shared__ int   sQ[TROWS];
  __shared__ float sNum[BLOCK];
  __shared__ float sCnt[BLOCK];

  const int tid = threadIdx.x;
  float accN = 0.0f;   // numerator partial
  float accC = 0.0f;   // non-pad row count partial

  const int numTiles = (rows + TROWS - 1) / TROWS;
  for (int t = blockIdx.x; t < numTiles; t += gridDim.x) {
    const int rowBase = t * TROWS;
    const long long elemBase = (long long)rowBase * NBINS;
    const int tileRows  = (rows - rowBase < TROWS) ? (rows - rowBase) : TROWS;
    const int tileElems = tileRows * NBINS;
    const int fullChunks = tileElems >> 2;        // 16B chunks

    // ---- async DMA: global -> LDS, 128b per lane, ASYNCcnt-tracked ----
    for (int c = tid; c < fullChunks; c += BLOCK) {
      unsigned ldsAddr = (unsigned)(unsigned long long)(const void*)&tile[c << 2];
      const float* gsrc = preds + elemBase + ((long long)c << 2);
      asm volatile("global_load_async_to_lds_b128 %0, %1, off"
                   :: "v"(ldsAddr), "v"(gsrc) : "memory");
    }
    for (int e = (fullChunks << 2) + tid; e < tileElems; e += BLOCK)
      tile[e] = preds[elemBase + e];              // tail (rare)

    // per-row quantized bin (-1 == pad / out of range); overlaps the DMA
    if (tid < TROWS) {
      int q = -1;
      int r = rowBase + tid;
      if (r < rows) {
        int g = gt[r];
        if (g != 100) {                           // PAD_F0
          int d = g - 50;                         // F_MIN
          q = (d <= 0) ? 0 : d / 6;               // SCALE, floor for d>=0
          if (q > NBINS - 1) q = NBINS - 1;
        }
      }
      sQ[tid] = q;
    }

    asm volatile("s_wait_asynccnt 0" ::: "memory");
    __syncthreads();

    // ---- Part A: softplus terms over all elements of non-pad rows ----
    for (int e = tid; e < tileElems; e += BLOCK) {
      int r = e / NBINS;
      if (sQ[r] >= 0) {
        float p = tile[e];
        accN += fmaxf(p, 0.0f) + __logf(1.0f + __expf(-fabsf(p)));
      }
    }

    // ---- Part B: subtract p . tgt (5-tap dot + reflect-pad corrections) ----
    if (tid < tileRows) {
      int q = sQ[tid];
      if (q >= 0) {
        const float* row = &tile[tid * NBINS];
        float dot = K2 * row[q];
        if (q >= 1)         dot += K1 * row[q - 1];
        if (q >= 2)         dot += K0 * row[q - 2];
        if (q <= NBINS - 2) dot += K1 * row[q + 1];
        if (q <= NBINS - 3) dot += K0 * row[q + 2];
        // reflect-padding extras (q in {1,2,47,48})
        if (q == 1)              dot += K0 * row[1] + K1 * row[0];
        else if (q == 2)         dot += K0 * row[0];
        else if (q == NBINS - 3) dot += K0 * row[NBINS - 1];
        else if (q == NBINS - 2) dot += K1 * row[NBINS - 1] + K0 * row[NBINS - 2];
        accN -= dot;
        accC += 1.0f;
      }
    }
    __syncthreads();   // before next tile's DMA overwrites LDS
  }

  float2 tot = block_reduce_pair(accN, accC, sNum, sCnt);
  if (tid == 0) {
    partials[2 * blockIdx.x]     = tot.x;
    partials[2 * blockIdx.x + 1] = tot.y;
  }
}

__global__ __launch_bounds__(BLOCK)
void pitch_final_kernel(const float* __restrict__ partials, int nblk,
                        float* __restrict__ out) {
  __shared__ float sNum[BLOCK];
  __shared__ float sCnt[BLOCK];
  const int tid = threadIdx.x;
  float accN = 0.0f, accC = 0.0f;
  for (int i = tid; i < nblk; i += BLOCK) {   // fixed order -> deterministic
    accN += partials[2 * i];
    accC += partials[2 * i + 1];
  }
  float2 tot = block_reduce_pair(accN, accC, sNum, sCnt);
  if (tid == 0) out[0] = tot.x / (50.0f * tot.y);
}

extern "C" void kernel_launch(void* const* d_in, const int* in_sizes, int n_in,
                              void* d_out, int out_size, void* d_ws, size_t ws_size,
                              hipStream_t stream) {
  (void)n_in; (void)out_size;
  const float* preds = (const float*)d_in[0];
  const int*   gt    = (const int*)d_in[1];
  float* out = (float*)d_out;
  float* ws  = (float*)d_ws;

  const int rows = in_sizes[1];                       // 128 * 4096
  const int numTiles = (rows + TROWS - 1) / TROWS;

  int nblk = MAXBLK;
  int wsCap = (int)(ws_size / (2 * sizeof(float)));
  if (nblk > wsCap)    nblk = wsCap;
  if (nblk > numTiles) nblk = numTiles;
  if (nblk < 1)        nblk = 1;

  pitch_partial_kernel<<<nblk, BLOCK, 0, stream>>>(preds, gt, ws, rows);
  pitch_final_kernel<<<1, BLOCK, 0, stream>>>(ws, nblk, out);
}